// LightGCN_5763846111287
// MI455X (gfx1250) — compile-verified
//
#include <hip/hip_runtime.h>
#include <stdint.h>

// ---------------- problem constants (match reference) ----------------
#define N_USER_   100000
#define N_ITEM_   200000
#define N_NODES_  (N_USER_ + N_ITEM_)
#define LATDIM_   64
#define NODE_F    ((size_t)N_NODES_ * LATDIM_)        // 19,200,000 floats
#define NODE_F4   (NODE_F / 4)                        //  4,800,000 float4

// ---------------- edge staging geometry ----------------
#define TILE            512                  // edges per LDS tile
#define NT              8                    // tiles per block
#define EDGES_PER_BLOCK (TILE * NT)          // 4096 edges per block

// Async global->LDS path (gfx1250, ASYNCcnt). Probe confirmed both builtins
// exist; signature is (v2i addrspace(1)*, v2i addrspace(3)*, imm, imm).
#if defined(__HIP_DEVICE_COMPILE__) && defined(__has_builtin)
#if __has_builtin(__builtin_amdgcn_global_load_async_to_lds_b64) && \
    __has_builtin(__builtin_amdgcn_s_wait_asynccnt)
#define USE_ASYNC_LDS 1
#endif
#endif
#ifndef USE_ASYNC_LDS
#define USE_ASYNC_LDS 0
#endif

#if defined(__HIP_DEVICE_COMPILE__)
typedef int lgcn_v2i __attribute__((vector_size(2 * sizeof(int))));
typedef __attribute__((address_space(1))) lgcn_v2i glb_v2i;   // global (AS1)
typedef __attribute__((address_space(3))) lgcn_v2i lds_v2i;   // LDS (AS3)
#endif

// ------------- stage one 512-edge tile of (row,col,val) into LDS ------------
// Every thread issues exactly 3 b64 async loads (2 edges x 3 arrays), so
// s_wait_asynccnt <= 3 is a precise "previous tile complete" condition.
// All edge indices fit in 31 bits (E = 9.6M) -> pure 32-bit address math.
__device__ __forceinline__ void lgcn_stage_tile(
    const int* __restrict__ erow, const int* __restrict__ ecol,
    const float* __restrict__ eval,
    int* sr, int* sc, float* sv,
    unsigned base, int tid, unsigned clampMax /* = nE-2 */)
{
    unsigned gi = base + 2u * (unsigned)tid;
    if (gi > clampMax) gi = 0u;             // clamp: always in-bounds, tail guarded later
#if USE_ASYNC_LDS
    __builtin_amdgcn_global_load_async_to_lds_b64(
        (glb_v2i*)(erow + gi), (lds_v2i*)(sr + 2 * tid), 0, 0);
    __builtin_amdgcn_global_load_async_to_lds_b64(
        (glb_v2i*)(ecol + gi), (lds_v2i*)(sc + 2 * tid), 0, 0);
    __builtin_amdgcn_global_load_async_to_lds_b64(
        (glb_v2i*)(eval + gi), (lds_v2i*)(sv + 2 * tid), 0, 0);
#else
    sr[2 * tid]     = erow[gi];
    sr[2 * tid + 1] = erow[gi + 1];
    sc[2 * tid]     = ecol[gi];
    sc[2 * tid + 1] = ecol[gi + 1];
    sv[2 * tid]     = eval[gi];
    sv[2 * tid + 1] = eval[gi + 1];
#endif
}

// ---------------- scatter SpMM: cout[row] += val * cin[col] -----------------
// wave32: one edge per wave iteration; lane owns dims [2*lane, 2*lane+1].
// Placed FIRST in the file so the disasm snippet shows this (hot) kernel.
__global__ __launch_bounds__(256) void lgcn_scatter_kernel(
    const int* __restrict__ erow, const int* __restrict__ ecol,
    const float* __restrict__ eval,
    const float* __restrict__ cin, float* __restrict__ cout, int nE)
{
    __shared__ __align__(16) int   s_row[2][TILE];
    __shared__ __align__(16) int   s_col[2][TILE];
    __shared__ __align__(16) float s_val[2][TILE];

    const int      tid   = threadIdx.x;
    const unsigned lane2 = 2u * (threadIdx.x & 31);   // lane's feature offset
    const int      wave  = tid >> 5;                  // 8 waves per block
    const unsigned bbase = (unsigned)blockIdx.x * EDGES_PER_BLOCK;
    const unsigned nEu   = (unsigned)nE;
    const unsigned cMax  = nEu - 2u;

    lgcn_stage_tile(erow, ecol, eval, s_row[0], s_col[0], s_val[0], bbase, tid, cMax);

    for (int t = 0; t < NT; ++t) {
        const int buf = t & 1;
        if (t + 1 < NT)
            lgcn_stage_tile(erow, ecol, eval,
                            s_row[(t + 1) & 1], s_col[(t + 1) & 1], s_val[(t + 1) & 1],
                            bbase + (unsigned)(t + 1) * TILE, tid, cMax);
#if USE_ASYNC_LDS
        // leave only the just-issued next tile (3 loads/thread) outstanding;
        // argument must be an integer constant expression.
        if (t + 1 < NT) __builtin_amdgcn_s_wait_asynccnt(3);
        else            __builtin_amdgcn_s_wait_asynccnt(0);
#endif
        __syncthreads();                    // tile t visible to all waves

        const unsigned tbase = bbase + (unsigned)t * TILE;
        for (int i = wave; i < TILE; i += 8) {
            if (tbase + (unsigned)i >= nEu) break;   // monotonic per wave -> safe
            const unsigned r = (unsigned)s_row[buf][i];
            const unsigned c = (unsigned)s_col[buf][i];
            const float    v = s_val[buf][i];

            // prefetch the gather row of this wave's next edge (global_prefetch_b8)
            const int inext = i + 8;
            if (inext < TILE && tbase + (unsigned)inext < nEu) {
                const unsigned cn = (unsigned)s_col[buf][inext];
                __builtin_prefetch(cin + ((cn << 6) + lane2), 0, 0);
            }

            // 32-bit offsets (max 19.2M) -> scalar-base + u32-voffset addressing
            const float2 x = *(const float2*)(cin + ((c << 6) + lane2));
            float* o = cout + ((r << 6) + lane2);
            unsafeAtomicAdd(o,     v * x.x);    // global_atomic_add_f32 (no return)
            unsafeAtomicAdd(o + 1, v * x.y);
        }
        __syncthreads();                    // all reads of buf done before restage
    }
}

// ---------------- init: acc = cur_in = concat(u,i); cur_out = 0 -------------
__global__ __launch_bounds__(256) void lgcn_init_kernel(
    const float* __restrict__ uE, const float* __restrict__ iE,
    float* __restrict__ acc, float* __restrict__ curA, float* __restrict__ curB)
{
    size_t idx = (size_t)blockIdx.x * blockDim.x + threadIdx.x;   // float4 index
    if (idx >= NODE_F4) return;
    const size_t u4 = (size_t)N_USER_ * LATDIM_ / 4;              // 1,600,000
    float4 v = (idx < u4) ? ((const float4*)uE)[idx]
                          : ((const float4*)iE)[idx - u4];
    ((float4*)acc )[idx] = v;
    ((float4*)curA)[idx] = v;
    ((float4*)curB)[idx] = make_float4(0.f, 0.f, 0.f, 0.f);
}

// -------- fused: acc += cur ; zero the buffer the next layer writes ---------
__global__ __launch_bounds__(256) void lgcn_accum_zero_kernel(
    float* __restrict__ acc, const float* __restrict__ cur, float* __restrict__ nextbuf)
{
    size_t idx = (size_t)blockIdx.x * blockDim.x + threadIdx.x;
    if (idx >= NODE_F4) return;
    float4 a = ((const float4*)acc)[idx];
    float4 c = ((const float4*)cur)[idx];
    a.x += c.x; a.y += c.y; a.z += c.z; a.w += c.w;
    ((float4*)acc)[idx]     = a;
    ((float4*)nextbuf)[idx] = make_float4(0.f, 0.f, 0.f, 0.f);
}

// --------------------------------- launch -----------------------------------
extern "C" void kernel_launch(void* const* d_in, const int* in_sizes, int n_in,
                              void* d_out, int out_size, void* d_ws, size_t ws_size,
                              hipStream_t stream)
{
    const float* uE = (const float*)d_in[0];
    const float* iE = (const float*)d_in[1];
    const int*   er = (const int*)  d_in[2];
    const int*   ec = (const int*)  d_in[3];
    const float* ev = (const float*)d_in[4];
    const int    nE = in_sizes[2];

    float* acc  = (float*)d_out;                   // [300000 x 64] = concat(u_out, i_out)
    float* curA = (float*)d_ws;                    // ping
    float* curB = curA + NODE_F;                   // pong (+76.8 MB)

    const int ewGrid = (int)((NODE_F4 + 255) / 256);               // 18750
    const int scGrid = (nE + EDGES_PER_BLOCK - 1) / EDGES_PER_BLOCK;

    // acc = cur = embeds ; curB = 0
    lgcn_init_kernel<<<ewGrid, 256, 0, stream>>>(uE, iE, acc, curA, curB);

    // layer 0: curB = A*curA ; acc += curB ; zero curA
    lgcn_scatter_kernel<<<scGrid, 256, 0, stream>>>(er, ec, ev, curA, curB, nE);
    lgcn_accum_zero_kernel<<<ewGrid, 256, 0, stream>>>(acc, curB, curA);

    // layer 1: curA = A*curB ; acc += curA ; zero curB
    lgcn_scatter_kernel<<<scGrid, 256, 0, stream>>>(er, ec, ev, curB, curA, nE);
    lgcn_accum_zero_kernel<<<ewGrid, 256, 0, stream>>>(acc, curA, curB);

    // layer 2: curB = A*curA ; acc += curB (zeroing curA is harmless)
    lgcn_scatter_kernel<<<scGrid, 256, 0, stream>>>(er, ec, ev, curA, curB, nE);
    lgcn_accum_zero_kernel<<<ewGrid, 256, 0, stream>>>(acc, curB, curA);
}